// DotProductPredictor_45887430591129
// MI455X (gfx1250) — compile-verified
//
#include <hip/hip_runtime.h>
#include <hip/hip_bf16.h>

typedef float v2f __attribute__((ext_vector_type(2)));
typedef float v8f __attribute__((ext_vector_type(8)));

#define D_FEAT 128

// Monotonic map: unsigned-compare order of key(f) == float order of f.
__device__ __forceinline__ unsigned fkey(float f) {
    unsigned b = __float_as_uint(f);
    return (b & 0x80000000u) ? ~b : (b | 0x80000000u);
}

__global__ void gnn_init_min(unsigned* __restrict__ minkey) {
    if (threadIdx.x == 0 && blockIdx.x == 0) *minkey = 0xFFFFFFFFu;  // +inf key
}

// One wave handles a tile of 16 edges. A = 16 gathered src rows (16xK),
// B = 16 gathered dst rows transposed (Kx16); diag(A*B) = 16 dot products.
// v_wmma_f32_16x16x4_f32 accumulates over K in steps of 4.
__global__ __launch_bounds__(256) void gnn_edge_dot_wmma(
    const float* __restrict__ h, const int* __restrict__ src,
    const int* __restrict__ dst, float* __restrict__ scores,
    unsigned* __restrict__ minkey, int n_edges) {
    const int lane = threadIdx.x & 31;
    const int wave = threadIdx.x >> 5;
    const int tile = blockIdx.x * 8 + wave;
    const int base = tile * 16;
    if (base >= n_edges) return;  // wave-uniform exit, EXEC stays full

    const int sub   = lane & 15;   // edge-in-tile for A rows / B cols
    const int khalf = lane >> 4;   // lanes 16-31 carry K+2,K+3 per ISA layout
    int e = base + sub;
    if (e >= n_edges) e = n_edges - 1;  // clamp tail; keeps EXEC all-ones for WMMA

    const float* srow = h + (size_t)src[e] * D_FEAT + 2 * khalf;
    const float* drow = h + (size_t)dst[e] * D_FEAT + 2 * khalf;

    v8f acc = {};
#pragma unroll
    for (int k = 0; k < D_FEAT; k += 4) {
        v2f a = *(const v2f*)(srow + k);  // A[M=sub][k+2*khalf .. +1]
        v2f b = *(const v2f*)(drow + k);  // B[k+2*khalf .. +1][N=sub]
        // 8 args: (neg_a, A, neg_b, B, c_mod, C, reuse_a, reuse_b)
        acc = __builtin_amdgcn_wmma_f32_16x16x4_f32(
            false, a, false, b, (short)0, acc, false, false);
    }

    // Diagonal extraction from 16x16 f32 C layout:
    // diag i (i<8)  lives in VGPR i,   lane i
    // diag i (i>=8) lives in VGPR i-8, lane i+16
    float sc = 0.0f;
#pragma unroll
    for (int v = 0; v < 8; ++v) {
        float d0 = __shfl(acc[v], v, 32);
        float d1 = __shfl(acc[v], v + 24, 32);
        if (lane == v) sc = d0;
        if (lane == v + 8) sc = d1;
    }

    const bool valid = (lane < 16) && ((base + lane) < n_edges);
    if (valid) scores[base + lane] = sc;

    // Wave-level min of monotonic keys, one atomic per wave.
    unsigned key = valid ? fkey(sc) : 0xFFFFFFFFu;
#pragma unroll
    for (int off = 16; off > 0; off >>= 1) {
        unsigned other = (unsigned)__shfl_xor((int)key, off, 32);
        key = (other < key) ? other : key;
    }
    if (lane == 0) atomicMin(minkey, key);
}

__global__ __launch_bounds__(256) void gnn_threshold(
    const float* __restrict__ scores, const unsigned* __restrict__ minkey,
    float* __restrict__ out, int n_edges) {
    int i = blockIdx.x * blockDim.x + threadIdx.x;
    if (i >= n_edges) return;
    unsigned k = *minkey;
    unsigned b = (k & 0x80000000u) ? (k ^ 0x80000000u) : ~k;  // inverse of fkey
    float mn = __uint_as_float(b);
    out[i] = (scores[i] == mn) ? 0.0f : 1.0f;
}

extern "C" void kernel_launch(void* const* d_in, const int* in_sizes, int n_in,
                              void* d_out, int out_size, void* d_ws, size_t ws_size,
                              hipStream_t stream) {
    const float* h  = (const float*)d_in[0];
    const int* src  = (const int*)d_in[1];
    const int* dst  = (const int*)d_in[2];
    float* out      = (float*)d_out;
    const int n_edges = in_sizes[1];

    // ws layout: [0,4) min key, scores start at byte 256 (aligned).
    unsigned* minkey = (unsigned*)d_ws;
    float* scores    = (float*)((char*)d_ws + 256);

    gnn_init_min<<<1, 1, 0, stream>>>(minkey);

    const int tiles  = (n_edges + 15) / 16;
    const int blocks = (tiles + 7) / 8;  // 8 waves (tiles) per 256-thread block
    gnn_edge_dot_wmma<<<blocks, 256, 0, stream>>>(h, src, dst, scores, minkey, n_edges);

    gnn_threshold<<<(n_edges + 255) / 256, 256, 0, stream>>>(scores, minkey, out, n_edges);
}